// PitchAutoCorrelator_73229192397002
// MI455X (gfx1250) — compile-verified
//
#include <hip/hip_runtime.h>

typedef __attribute__((ext_vector_type(2))) float v2f;
typedef __attribute__((ext_vector_type(4))) float v4f;
typedef __attribute__((ext_vector_type(8))) float v8f;

#define FRAME   80
#define WIN     5
#define BATCH   64
#define NFRAMES 4000
#define NSAMP   (FRAME * NFRAMES)               // 320000
#define GPAIRS  8                               // frame-pairs per wave
#define GRPB    (NFRAMES / (2 * GPAIRS))        // 250 groups per batch
#define GROUPS  (BATCH * GRPB)                  // 16000 waves
#define STAGE   1664                            // staged floats (13 b128/lane)
#define FILL    1584                            // slow-path fill extent
#define ZOFF    1664                            // zero area (80 floats, copy A)
#define CB      1744                            // copy B base (float units)
#define WREG    3424                            // floats per wave region
#define CBOFF   ((CB + 1) * 4)                  // byte delta for odd-parity reads

// One wave32 per group of 8 frame-pairs. Per pair, A (16 x 80):
//   m=0: frame fA; m=1..5: lags of fA; m=6,7: zero
//   m=8: frame fB; m=9..13: lags of fB; m=14,15: zero
// G = A*A^T via 20 chained V_WMMA_F32_16X16X4_F32 (A-fragment == B-fragment).
// LDS holds the span twice (copy B shifted by one float) so every fragment
// pair is 8B-aligned -> DS_LOAD_2ADDR_B64 fetches two K-steps per instruction.
__global__ __launch_bounds__(128, 1) void pitch_acorr_wmma(
    const float* __restrict__ x,
    const int*   __restrict__ periods,
    float*       __restrict__ out)
{
    __shared__ float sm[4 * WREG];

    const int lane = threadIdx.x & 31;
    const int wv   = threadIdx.x >> 5;
    const int wave = blockIdx.x * 4 + wv;

    const int b  = wave / GRPB;
    const int g  = wave % GRPB;
    const int f0 = g * 2 * GPAIRS;

    const float* __restrict__ xb = x + (size_t)b * NSAMP;
    const int gLo = f0 * FRAME - 300;          // multiple of 4 -> 16B aligned
    float* wsm = sm + wv * WREG;               // private per-wave region

    // zero area for the dead rows (80 floats) in copy A coords
    if (lane < 20) *(v4f*)(wsm + ZOFF + lane * 4) = (v4f){0.f, 0.f, 0.f, 0.f};

    const int sIdx = ((lane + 31) & 31) * 4;   // bpermute: lane <- lane-1

    if (gLo >= 0 && gLo + (STAGE - 1) < NSAMP) {
        // fast staging: copy A verbatim, copy B shifted by one element
        const float* src = xb + gLo;
        #pragma unroll
        for (int c = 0; c < STAGE / 128; ++c) {
            v4f val = *(const v4f*)(src + lane * 4 + c * 128);
            *(v4f*)(wsm + lane * 4 + c * 128) = val;            // copy A
            float pw = __int_as_float(
                __builtin_amdgcn_ds_bpermute(sIdx, __float_as_int(val.w)));
            v4f sh;
            sh.x = pw; sh.y = val.x; sh.z = val.y; sh.w = val.z;
            *(v4f*)(wsm + CB + lane * 4 + c * 128) = sh;        // copy B
        }
        // chunk-boundary cleanup: copy B slot 128c must hold element 128c-1
        if (lane >= 1 && lane <= 12)
            wsm[CB + 128 * lane] = wsm[128 * lane - 1];
    } else {
        // first/last group per batch only (128/16000 waves): JAX gather rules
        for (int i = lane; i < FILL; i += 32) {
            const int idx = gLo + i;
            float v = 0.0f;
            if (idx >= 0) v = xb[idx < NSAMP ? idx : NSAMP - 1];
            wsm[i] = v;
        }
        for (int s = lane + 1; s < FILL; s += 32)
            wsm[CB + s] = wsm[s - 1];
    }

    // ---- per-lane constants (hoisted) ----
    const int m  = lane & 15;
    const int hi = lane >> 4;
    const bool zeroRow = (m == 6) | (m == 7) | (m == 14) | (m == 15);
    const int  mm = m & 7;
    const bool isLag   = (mm != 0) && !zeroRow;
    const int  lagMask = isLag ? ~0 : 0;
    const int  lagAdj  = mm - 3;
    const int* __restrict__ pb = periods + b * NFRAMES + f0 + ((m < 8) ? 0 : 1);

    // preload all 8 periods (one clause, imm offsets, no per-pair addr math)
    int pv[GPAIRS];
    #pragma unroll
    for (int i = 0; i < GPAIRS; ++i) pv[i] = pb[2 * i];

    int relBase = zeroRow ? ZOFF : (((m < 8) ? 300 : 380) + 2 * hi);
    const int step = zeroRow ? 0 : 160;        // floats advanced per pair

    // epilogue invariants
    const int  li  = (lane < 16) ? lane : lane - 24;     // 1..5 = active
    const bool a2 = (li == 2), a3 = (li == 3), a4 = (li == 4), a5 = (li == 5);
    const bool act = (li >= 1) && (li <= 5);
    int permIdx = (lane < 16) ? 0 : 96;                  // byte idx: lane 0 / 24
    asm("" : "+v"(permIdx));                             // pin: no remat per pair

    float res[GPAIRS];

    #pragma unroll
    for (int i = 0; i < GPAIRS; ++i) {
        const int r    = relBase + ((lagAdj - pv[i]) & lagMask);
        const int addr = (r << 2) + ((r & 1) ? CBOFF : 0);   // parity select
        const v2f* __restrict__ bp = (const v2f*)((const char*)wsm + addr);
        relBase += step;

        v8f acc = {};
        #pragma unroll
        for (int t = 0; t < 10; ++t) {
            v2f a0 = bp[4 * t];        // -> ds_load_2addr_b64 offs 4t, 4t+2
            v2f a1 = bp[4 * t + 2];
            acc = __builtin_amdgcn_wmma_f32_16x16x4_f32(
                      false, a0, false, a0, (short)0, acc, false, false);
            acc = __builtin_amdgcn_wmma_f32_16x16x4_f32(
                      false, a1, false, a1, (short)0, acc, false, false);
        }

        // D layout: lanes 0-15 cols N=lane rows 0-7 in v0-7; lanes 16-31
        // cols N=lane-16 rows 8-15.
        const float fnrg = __int_as_float(
            __builtin_amdgcn_ds_bpermute(permIdx, __float_as_int(acc[0])));

        float lagn = acc[1];
        lagn = a2 ? acc[2] : lagn;
        lagn = a3 ? acc[3] : lagn;
        lagn = a4 ? acc[4] : lagn;
        lagn = a5 ? acc[5] : lagn;

        res[i] = acc[0] * __builtin_amdgcn_rsqf(fnrg * lagn + 1e-9f);
    }

    // single guarded store block: 8 stores with pure immediate offsets
    if (act) {
        float* __restrict__ outp = out +
            ((size_t)b * NFRAMES + f0 + ((lane < 16) ? 0 : 1)) * WIN + (li - 1);
        #pragma unroll
        for (int i = 0; i < GPAIRS; ++i)
            outp[i * 2 * WIN] = res[i];        // imm offset: i*40 bytes
    }
}

extern "C" void kernel_launch(void* const* d_in, const int* in_sizes, int n_in,
                              void* d_out, int out_size, void* d_ws, size_t ws_size,
                              hipStream_t stream)
{
    const float* x       = (const float*)d_in[0];   // (64, 1, 320000) f32
    const int*   periods = (const int*)  d_in[1];   // (64, 4000) i32
    float*       out     = (float*)d_out;           // (64, 1, 4000, 5) f32

    const int threads = 128;             // 4 waves per block
    const int blocks  = GROUPS / 4;      // exact: 4000 blocks
    pitch_acorr_wmma<<<blocks, threads, 0, stream>>>(x, periods, out);
}